// LorenzModel_80762565034291
// MI455X (gfx1250) — compile-verified
//
#include <hip/hip_runtime.h>
#include <hip/hip_bf16.h>
#include <stdint.h>

// Lorenz Euler integration: strictly serial 3-state recurrence, 3999 steps.
// Latency-bound on one dependency chain -> single wave32, lane 0 integrates in
// registers, trajectory staged in LDS via ds_store, then ONE Tensor Data Mover
// DMA (tensor_store_from_lds, TENSORcnt) writes the whole 48KB trajectory to
// global memory.

#define LORENZ_DT 0.01f
#define MAX_ELEMS 12000   // 4000 steps * 3 components (fp32) = 48,000 B of LDS

typedef __attribute__((ext_vector_type(4))) unsigned int u32x4;
typedef __attribute__((ext_vector_type(8))) int          i32x8;
typedef __attribute__((ext_vector_type(4))) int          i32x4;

__global__ __launch_bounds__(32) void lorenz_tdm_kernel(
    const float* __restrict__ sigma,
    const float* __restrict__ rho,
    const float* __restrict__ beta,
    const float* __restrict__ stats,
    float* __restrict__ out,
    int T)
{
    __shared__ float traj[MAX_ELEMS];

    // Clamp for safety against the static LDS allocation.
    if (3 * T > MAX_ELEMS) T = MAX_ELEMS / 3;

    const float s = sigma[0];
    const float r = rho[0];
    const float b = beta[0];

    float x = stats[0];
    float y = stats[1];
    float z = stats[2];

    // Serial integration on lane 0 only; per-step results parked in LDS
    // (ds_store is fire-and-forget on DScnt; the dependency chain never
    // waits on memory).
    if (threadIdx.x == 0) {
        traj[0] = x; traj[1] = y; traj[2] = z;
        for (int i = 1; i < T; ++i) {
            const float dx = s * (y - x);
            const float dy = x * (r - z) - y;
            const float dz = x * y - b * z;
            x += LORENZ_DT * dx;
            y += LORENZ_DT * dy;
            z += LORENZ_DT * dz;
            traj[3 * i + 0] = x;
            traj[3 * i + 1] = y;
            traj[3 * i + 2] = z;
        }
    }

    // Make LDS contents visible before the TDM engine reads them.
    asm volatile("s_wait_dscnt 0" ::: "memory");

    // ---- Tensor DMA descriptor (D#), CDNA5 ISA ----
    // 1-D tensor: tensor_dim0 = tile_dim0 = 3*T fp32 elements, dim1 = 1.
    const unsigned n = (unsigned)(3 * T);

    // addrspace(3) offset: for the LDS aperture the low 32 bits of the flat
    // pointer are the LDS byte address (ISA 10.2 aperture mapping).
    const unsigned lds_addr = (unsigned)(uintptr_t)(&traj[0]);
    const unsigned long long gaddr = (unsigned long long)(uintptr_t)out;

    u32x4 g0;
    g0[0] = 1u;                                   // [1:0] count=1 (valid), user mode
    g0[1] = lds_addr;                             // [63:32] lds_addr
    g0[2] = (unsigned)(gaddr & 0xFFFFFFFFu);      // [95:64] global_addr lo
    g0[3] = (unsigned)((gaddr >> 32) & 0x1FFFFFFu)// [120:96] global_addr hi
          | (2u << 30);                           // [127:126] type = 2 ("image")

    i32x8 g1;
    g1[0] = (int)(2u << 16);                      // data_size = 2 (4 bytes); wg_mask=0
    g1[1] = (int)((n & 0xFFFFu) << 16);           // tensor_dim0[15:0] at [63:48]
    g1[2] = (int)(((n >> 16) & 0xFFFFu)           // tensor_dim0[31:16]
          | (1u << 16));                          // tensor_dim1 = 1
    g1[3] = (int)((n & 0xFFFFu) << 16);           // tile_dim0 = n (tensor_dim1 hi = 0)
    g1[4] = 1;                                    // tile_dim1 = 1, tile_dim2 = 0
    g1[5] = (int)n;                               // tensor_dim0_stride[31:0]
    g1[6] = (int)((n & 0xFFFFu) << 16);           // stride0 hi = 0 | stride1[15:0]
    g1[7] = (int)((n >> 16) & 0xFFFFu);           // stride1[47:16]

    i32x4 g2 = {0, 0, 0, 0};                      // unused (<= 2-D tensor)
    i32x4 g3 = {0, 0, 0, 0};
    i32x8 g4 = {0, 0, 0, 0, 0, 0, 0, 0};          // extra group, 6-arg builtin form

    // One DMA: LDS -> global, entire trajectory. clang-23 6-arg form.
    __builtin_amdgcn_tensor_store_from_lds(g0, g1, g2, g3, g4, /*cpol=*/0);

    // Drain TENSORcnt before wave exit.
    __builtin_amdgcn_s_wait_tensorcnt(0);
}

extern "C" void kernel_launch(void* const* d_in, const int* in_sizes, int n_in,
                              void* d_out, int out_size, void* d_ws, size_t ws_size,
                              hipStream_t stream) {
    // setup_inputs order: t[4000], sigma[1], rho[1], beta[1], stats[3]
    const float* sigma = (const float*)d_in[1];
    const float* rho   = (const float*)d_in[2];
    const float* beta  = (const float*)d_in[3];
    const float* stats = (const float*)d_in[4];
    const int T = in_sizes[0];                    // 4000 time points

    lorenz_tdm_kernel<<<1, 32, 0, stream>>>(sigma, rho, beta, stats,
                                            (float*)d_out, T);
}